// SPS_867583394309
// MI455X (gfx1250) — compile-verified
//
#include <hip/hip_runtime.h>

// CDNA5 / gfx1250: wave32, WMMA f32 16x16x4 used for the channel reduction.
typedef float v2f __attribute__((ext_vector_type(2)));
typedef float v8f __attribute__((ext_vector_type(8)));

namespace {
constexpr int kC = 5000;          // channels
constexpr int kL = 4000;          // sequence length of x
constexpr int kNWaves = 10;       // waves per block
constexpr int kBlock = kNWaves * 32;   // 320 threads
constexpr int kNChunk = kC / 4;        // 1250 K-chunks of 4 channels
constexpr int kPerWave = kNChunk / kNWaves;  // 125, exact -> constant trip count
}

__global__ __launch_bounds__(kBlock)
void sps_fused_wmma_867583394309(
    const float* __restrict__ x,   // [8, 5000, 4000]
    const float* __restrict__ w,   // [5000]
    const float* __restrict__ g1, const float* __restrict__ b1,
    const float* __restrict__ m1, const float* __restrict__ v1,
    const float* __restrict__ wr,
    const float* __restrict__ g2, const float* __restrict__ b2,
    const float* __restrict__ m2, const float* __restrict__ v2,
    float* __restrict__ out)       // [8, 500] flat = 4000
{
    __shared__ __align__(16) float s_w[kC];      // 20KB: w_proj staged in LDS
    __shared__ float s_red[kNWaves * 16];        // per-wave partial sums
    __shared__ float s_z[16];                    // post-BN2 values before pooling

    const int tid = threadIdx.x;

    // Cooperative staging of w_proj into LDS (b128 stores).
    {
        const float4* w4 = (const float4*)w;
        float4* s4 = (float4*)s_w;
        for (int i = tid; i < kC / 4; i += kBlock) s4[i] = w4[i];
    }
    __syncthreads();

    const int wave = tid >> 5;
    const int lane = tid & 31;
    const int m    = lane & 15;   // A-matrix row index M
    const int hi   = lane >> 4;   // 0: K0/K1 half, 1: K2/K3 half (ISA A layout)

    // Flat output-row index n = 16*blockIdx.x + m in [0, 8000): b = n/1000, e = n%1000.
    // Only even y positions are live: x column = 4*e.
    const int n = blockIdx.x * 16 + m;
    const int b = n / 1000;
    const int e = n - b * 1000;
    const float* xrow = x + (size_t)b * (size_t)kC * (size_t)kL + (size_t)(4 * e);

    v8f acc = {0.f, 0.f, 0.f, 0.f, 0.f, 0.f, 0.f, 0.f};

    // Wave 'wave' owns the contiguous chunk range [wave*125, (wave+1)*125):
    // constant trip count (125 = 5*25) -> clean scalar loop, unrollable;
    // global base advances 4*kL floats (64000B) per iter, LDS base 16B per iter,
    // both reachable with immediate offsets inside the unrolled body.
    const int kbase = wave * kPerWave;
    const float* p = xrow + (size_t)(4 * kbase + 2 * hi) * (size_t)kL;
    const v2f* wp = ((const v2f*)s_w) + (2 * kbase + hi);

    #pragma unroll 5
    for (int i = 0; i < kPerWave; ++i) {
        // B operand: w[c0+2hi], w[c0+2hi+1] broadcast (every D column equal).
        v2f wb = wp[2 * i];
        // A operand: x[b, c0+2hi + {0,1}, 4e] -- pure stream, non-temporal.
        float a0 = __builtin_nontemporal_load(p);
        float a1 = __builtin_nontemporal_load(p + kL);
        v2f av = {a0, a1};
        // D = A(16x4,f32) * B(4x16,f32) + D  -- exact fp32 accumulation
        acc = __builtin_amdgcn_wmma_f32_16x16x4_f32(
            false, av, false, wb, (short)0, acc, false, false);
        p += (size_t)(4 * kL);
    }

    // D layout: VGPR i @ lane0  -> (M=i,   N=0);  VGPR i @ lane16 -> (M=8+i, N=0).
    if (lane == 0) {
        #pragma unroll
        for (int i = 0; i < 8; ++i) s_red[wave * 16 + i] = acc[i];
    } else if (lane == 16) {
        #pragma unroll
        for (int i = 0; i < 8; ++i) s_red[wave * 16 + 8 + i] = acc[i];
    }
    __syncthreads();

    // Cross-wave reduction + BN1 + spike + rpe scale + BN2.
    if (tid < 16) {
        float y = 0.f;
        #pragma unroll
        for (int wv = 0; wv < kNWaves; ++wv) y += s_red[wv * 16 + tid];
        const float k1 = g1[0] * rsqrtf(v1[0] + 1e-5f);
        y = (y - m1[0]) * k1 + b1[0];
        const float s = (y / 1.1f >= 1.0f) ? 1.0f : 0.0f;   // proj_lif spike
        float z = wr[0] * s;                                 // rpe_conv (k=1, stride2 folded)
        const float k2 = g2[0] * rsqrtf(v2[0] + 1e-5f);
        z = (z - m2[0]) * k2 + b2[0];
        s_z[tid] = z;
    }
    __syncthreads();

    // AdaptiveAvgPool(2) + final spike; 8 outputs per block.
    if (tid < 8) {
        const float zp = 0.5f * (s_z[2 * tid] + s_z[2 * tid + 1]);
        out[blockIdx.x * 8 + tid] = (zp / 1.1f >= 1.0f) ? 1.0f : 0.0f;
    }
}

extern "C" void kernel_launch(void* const* d_in, const int* in_sizes, int n_in,
                              void* d_out, int out_size, void* d_ws, size_t ws_size,
                              hipStream_t stream) {
    (void)in_sizes; (void)n_in; (void)d_ws; (void)ws_size; (void)out_size;
    const float* x  = (const float*)d_in[0];
    const float* w  = (const float*)d_in[1];
    const float* g1 = (const float*)d_in[2];
    const float* b1 = (const float*)d_in[3];
    const float* m1 = (const float*)d_in[4];
    const float* v1 = (const float*)d_in[5];
    const float* wr = (const float*)d_in[6];
    const float* g2 = (const float*)d_in[7];
    const float* b2 = (const float*)d_in[8];
    const float* m2 = (const float*)d_in[9];
    const float* v2 = (const float*)d_in[10];
    float* out = (float*)d_out;

    // 8000 live (b,e) rows / 16 per block = 500 blocks; 8 outputs each = 4000.
    dim3 grid(500), block(kBlock);
    hipLaunchKernelGGL(sps_fused_wmma_867583394309, grid, block, 0, stream,
                       x, w, g1, b1, m1, v1, wr, g2, b2, m2, v2, out);
}